// TransferEntropyLoss_77103252898289
// MI455X (gfx1250) — compile-verified
//
#include <hip/hip_runtime.h>
#include <hip/hip_bf16.h>

#ifndef __has_builtin
#define __has_builtin(x) 0
#endif

#define N_SAMP 1024
#define T_DIM 8
#define V_DIM 4
#define D_DIM 8
#define M_ROWS 1020          // N - HIST_K
#define P_TASKS 28           // V * (T-1)
#define NW 8                 // waves per block (256 threads, wave32)
#define ROWBLKS 128          // 128 * 8 = 1024 rows covered (1020 real + 4 zero-pad)
#define TERM_STRIDE 1024     // padded per-task stride -> 28*1024 = 448 full WMMA chunks
#define N_TERMS (P_TASKS * TERM_STRIDE)

typedef __attribute__((ext_vector_type(2))) float v2f;
typedef __attribute__((ext_vector_type(8))) float v8f;

// ---------------- device helpers ----------------

__device__ __forceinline__ float cheb8(const float4 a0, const float4 a1,
                                       const float4* __restrict__ b) {
  const float4 b0 = b[0], b1 = b[1];
  float m = fabsf(a0.x - b0.x);
  m = fmaxf(m, fabsf(a0.y - b0.y));
  m = fmaxf(m, fabsf(a0.z - b0.z));
  m = fmaxf(m, fabsf(a0.w - b0.w));
  m = fmaxf(m, fabsf(a1.x - b1.x));
  m = fmaxf(m, fabsf(a1.y - b1.y));
  m = fmaxf(m, fabsf(a1.z - b1.z));
  m = fmaxf(m, fabsf(a1.w - b1.w));
  return m;
}

__device__ __forceinline__ void ins3(float& s0, float& s1, float& s2, float v) {
  if (v < s2) {
    if (v < s1) {
      s2 = s1;
      if (v < s0) { s1 = s0; s0 = v; } else { s1 = v; }
    } else {
      s2 = v;
    }
  }
}

__device__ __forceinline__ float digammaf(float x) {
  // psi(x) for x >= 1 (here x = count+1, integer-valued float)
  float r = 0.f;
  while (x < 6.f) { r -= 1.f / x; x += 1.f; }
  const float inv  = 1.f / x;
  const float inv2 = inv * inv;
  r += __logf(x) - 0.5f * inv
     - inv2 * (0.08333333333f - inv2 * (0.00833333333f - inv2 * 0.00396825397f));
  return r;
}

struct RowRegs {
  float4 ta[5][2];   // tgt[i..i+4]
  float4 sa[4][2];   // src[i..i+3]
};

__device__ __forceinline__ void pair_dists(const RowRegs& r,
                                           const float4* __restrict__ tg,
                                           const float4* __restrict__ sg,
                                           int j, float& dA, float& dB, float& dC) {
  dA = cheb8(r.ta[4][0], r.ta[4][1], &tg[(j + 4) * 2]);
  dC = cheb8(r.ta[0][0], r.ta[0][1], &tg[j * 2]);
  dB = cheb8(r.sa[0][0], r.sa[0][1], &sg[j * 2]);
#pragma unroll
  for (int l = 1; l < 4; ++l) {
    dC = fmaxf(dC, cheb8(r.ta[l][0], r.ta[l][1], &tg[(j + l) * 2]));
    dB = fmaxf(dB, cheb8(r.sa[l][0], r.sa[l][1], &sg[(j + l) * 2]));
  }
}

// ---------------- main kernel: one TE task per blockIdx.x, 8 rows per block ----------------

__global__ __launch_bounds__(256) void te_rows_kernel(const float* __restrict__ x,
                                                      float* __restrict__ terms) {
  __shared__ float sh_src[N_SAMP * D_DIM];   // 32 KB
  __shared__ float sh_tgt[N_SAMP * D_DIM];   // 32 KB

  const int p = blockIdx.x;                 // 0..27
  const int v = p / (T_DIM - 1);
  const int t = p % (T_DIM - 1) + 1;

  // x layout: [b][t][v][d] flat stride 256/32/8/1
  const float* srcBase = x + t * (V_DIM * D_DIM) + v * D_DIM;
  const float* tgtBase = x + v * D_DIM;

  // Stage src & tgt series into LDS via CDNA5 async global->LDS (ASYNCcnt path).
  // 4096 float4 transfers total (2048 src + 2048 tgt), 16 per thread.
  for (int k = threadIdx.x; k < 4096; k += 256) {
    const int which = k >> 11;              // 0 = src, 1 = tgt
    const int kk    = k & 2047;
    const int n     = kk >> 1;              // sample index
    const int half  = kk & 1;               // which float4 of the 8-dim row
    const float* g  = (which ? tgtBase : srcBase) + n * (T_DIM * V_DIM * D_DIM) + half * 4;
    float* l        = (which ? sh_tgt : sh_src) + kk * 4;
    const unsigned int       laddr = (unsigned int)(unsigned long long)l;
    const unsigned long long gaddr = (unsigned long long)g;
    asm volatile("global_load_async_to_lds_b128 %0, %1, off"
                 :: "v"(laddr), "v"(gaddr) : "memory");
  }
  asm volatile("s_wait_asynccnt 0x0" ::: "memory");
  __syncthreads();

  const int wave = threadIdx.x >> 5;
  const int lane = threadIdx.x & 31;
  const int i    = blockIdx.y * NW + wave;

  if (i >= M_ROWS) {
    // zero-fill padding rows 1020..1023 so the WMMA reduction needs no tail handling
    if (lane == 0 && i < TERM_STRIDE) terms[p * TERM_STRIDE + i] = 0.f;
    return;
  }

  const float4* tg = (const float4*)sh_tgt;
  const float4* sg = (const float4*)sh_src;

  RowRegs r;
#pragma unroll
  for (int l = 0; l < 5; ++l) { r.ta[l][0] = tg[(i + l) * 2]; r.ta[l][1] = tg[(i + l) * 2 + 1]; }
#pragma unroll
  for (int l = 0; l < 4; ++l) { r.sa[l][0] = sg[(i + l) * 2]; r.sa[l][1] = sg[(i + l) * 2 + 1]; }

  // ---- pass 1: 3 smallest joint distances (k=3 NN radius) ----
  const float INFV = __int_as_float(0x7f800000);
  float s0 = INFV, s1 = INFV, s2 = INFV;
  for (int j = lane; j < M_ROWS; j += 32) {
    float dA, dB, dC;
    pair_dists(r, tg, sg, j, dA, dB, dC);
    const float dJ = fmaxf(fmaxf(dA, dC), dB);
    if (j != i) ins3(s0, s1, s2, dJ);
  }
  // duplicate-free hypercube merge of per-lane 3-smallest sets
#pragma unroll
  for (int off = 16; off >= 1; off >>= 1) {
    const float t0 = __shfl_xor(s0, off, 32);
    const float t1 = __shfl_xor(s1, off, 32);
    const float t2 = __shfl_xor(s2, off, 32);
    ins3(s0, s1, s2, t0);
    ins3(s0, s1, s2, t1);
    ins3(s0, s1, s2, t2);
  }
  const float eps = s2;   // 3rd smallest (KNN_K = 3)

  // ---- pass 2: strict-< neighbor counts in marginal spaces ----
  int nAC = 0, nBC = 0, nC = 0;
  for (int j = lane; j < M_ROWS; j += 32) {
    if (j == i) continue;
    float dA, dB, dC;
    pair_dists(r, tg, sg, j, dA, dB, dC);
    nAC += (fmaxf(dA, dC) < eps) ? 1 : 0;
    nBC += (fmaxf(dB, dC) < eps) ? 1 : 0;
    nC  += (dC < eps) ? 1 : 0;
  }
#pragma unroll
  for (int off = 16; off >= 1; off >>= 1) {
    nAC += __shfl_xor(nAC, off, 32);
    nBC += __shfl_xor(nBC, off, 32);
    nC  += __shfl_xor(nC,  off, 32);
  }

  if (lane == 0) {
    const float term = digammaf((float)nC + 1.f)
                     - digammaf((float)nAC + 1.f)
                     - digammaf((float)nBC + 1.f);
    terms[p * TERM_STRIDE + i] = term;
  }
}

// ---------------- reduction: sum 28*1024 padded terms with V_WMMA_F32_16X16X4_F32 ----------------

__global__ __launch_bounds__(32) void te_reduce_kernel(const float* __restrict__ terms,
                                                       float* __restrict__ out) {
  const int lane = threadIdx.x;
  float vsum;
#if __has_builtin(__builtin_amdgcn_wmma_f32_16x16x4_f32)
  // D = ones(16x4) * B(4x16) + C : every D row = column sums of B; accumulate over chunks.
  // N_TERMS = 28672 = 448 * 64 exactly -> no bounds checks, unconditional coalesced loads.
  v2f a; a.x = 1.f; a.y = 1.f;                // A all-ones (16x4 f32, 2 VGPRs/lane)
  v8f c = {0.f, 0.f, 0.f, 0.f, 0.f, 0.f, 0.f, 0.f};
  const int nidx  = lane & 15;                // B: lanes 0-15 -> K0,K1 ; lanes 16-31 -> K2,K3
  const int khalf = lane >> 4;
  const float* base0 = terms + (khalf * 2 + 0) * 16 + nidx;
  const float* base1 = terms + (khalf * 2 + 1) * 16 + nidx;
  for (int ch = 0; ch < N_TERMS / 64; ++ch) {
    v2f b;
    b.x = base0[ch * 64];
    b.y = base1[ch * 64];
    c = __builtin_amdgcn_wmma_f32_16x16x4_f32(false, a, false, b, (short)0, c,
                                              false, false);
  }
  // Row M=0 of D lives in c[0], lanes 0..15 (lanes 16-31 hold duplicate row M=8 -> drop)
  vsum = (lane < 16) ? c[0] : 0.f;
#else
  vsum = 0.f;
  for (int idx = lane; idx < N_TERMS; idx += 32) vsum += terms[idx];
#endif
#pragma unroll
  for (int off = 16; off >= 1; off >>= 1) vsum += __shfl_xor(vsum, off, 32);

  if (lane == 0) {
    const float PSI3  = 0.92278433509846714f;       // digamma(3)
    const float SCALE = 0.1f / 256.0f;              // BETA / (T*V*D)
    // mean over aggs = 7*psi(3) + S/(4*M)
    out[0] = SCALE * (7.0f * PSI3 + vsum / (4.0f * (float)M_ROWS));
  }
}

// ---------------- launch ----------------

extern "C" void kernel_launch(void* const* d_in, const int* in_sizes, int n_in,
                              void* d_out, int out_size, void* d_ws, size_t ws_size,
                              hipStream_t stream) {
  (void)in_sizes; (void)n_in; (void)out_size; (void)ws_size;
  const float* x = (const float*)d_in[0];
  float* terms   = (float*)d_ws;                    // 28*1024 floats = 112 KB scratch

  dim3 grid(P_TASKS, ROWBLKS, 1);
  te_rows_kernel<<<grid, 256, 0, stream>>>(x, terms);
  te_reduce_kernel<<<1, 32, 0, stream>>>(terms, (float*)d_out);
}